// Renderer_5823975653813
// MI455X (gfx1250) — compile-verified
//
#include <hip/hip_runtime.h>
#include <hip/hip_bf16.h>
#include <stdint.h>

// Problem geometry (fixed by the reference).
#define HW_PIX   (1024 * 1024)   // H*W pixels
#define D_ATTR   16              // attribute dim
#define TILE     64              // pixels per block-tile (256 thr / 4 lanes-per-pixel)
#define NTILES   (HW_PIX / TILE) // 16384
#define NBLOCKS  4096
#define NTHREADS 256

// Native 4-float vector (clang ext vector) -- accepted by the nontemporal
// builtins, loads/stores as one b128.
typedef float v4f __attribute__((ext_vector_type(4)));

// ---------------------------------------------------------------------------
// CDNA5 async global->LDS copies (ASYNCcnt-tracked, no VGPR destination).
// GVS addressing: 64-bit SGPR base + 32-bit per-lane VGPR byte offset.
// LDS destination byte address is carried in a VGPR (wave-relative offset =
// low 32 bits of the generic pointer to a __shared__ object; verified to
// constant-fold in round 1's disassembly).
// ---------------------------------------------------------------------------
__device__ __forceinline__ void async_ld_b64(uint32_t lds_addr, uint32_t voff,
                                             const void* sbase) {
  asm volatile("global_load_async_to_lds_b64 %0, %1, %2 offset:0"
               :: "v"(lds_addr), "v"(voff), "s"(sbase) : "memory");
}
__device__ __forceinline__ void async_ld_b128(uint32_t lds_addr, uint32_t voff,
                                              const void* sbase) {
  asm volatile("global_load_async_to_lds_b128 %0, %1, %2 offset:0"
               :: "v"(lds_addr), "v"(voff), "s"(sbase) : "memory");
}
__device__ __forceinline__ void wait_async_le1() {
  asm volatile("s_wait_asynccnt 0x1" ::: "memory");
}
__device__ __forceinline__ void wait_async_0() {
  asm volatile("s_wait_asynccnt 0x0" ::: "memory");
}

// Stage one 64-pixel tile's streaming inputs into LDS.
//  threads [0,64)   : pix_to_face, one b64 each            (512 B)
//  threads [64,112) : bary_coords, one b128 each (48*16B = 768 B = 64*12 B)
// Exactly one async instruction per issuing wave per tile -> the per-wave
// ASYNCcnt bookkeeping below is exact (<=1 == previous tile's batch landed).
__device__ __forceinline__ void stage_tile(int t, uint32_t pix0,
                                           uint32_t ptf_lds, uint32_t bary_lds,
                                           const long long* ptf,
                                           const float* bary) {
  if (t < TILE) {
    async_ld_b64(ptf_lds + (uint32_t)t * 8u, (pix0 + (uint32_t)t) * 8u, ptf);
  } else if (t < TILE + 48) {
    uint32_t j = (uint32_t)(t - TILE);
    async_ld_b128(bary_lds + j * 16u, pix0 * 12u + j * 16u, bary);
  }
}

__global__ void __launch_bounds__(NTHREADS)
renderer_interp_kernel(const float* __restrict__ attributes,   // (V,16) f32
                       const float* __restrict__ bary,         // (HW,3) f32
                       const long long* __restrict__ faces,    // (F,3) i64
                       const long long* __restrict__ ptf,      // (HW,) i64
                       float* __restrict__ out,                // (HW,16) f32
                       float* __restrict__ mask)               // (HW,) f32
{
  __shared__ __align__(16) float     s_bary[2][TILE * 3];
  __shared__ __align__(8)  long long s_ptf[2][TILE];

  const int t = (int)threadIdx.x;
  const int p = t >> 2;   // local pixel 0..63
  const int c = t & 3;    // float4 chunk 0..3 of D=16

  const uint32_t bary_lds0 = (uint32_t)(uintptr_t)&s_bary[0][0];
  const uint32_t bary_lds1 = (uint32_t)(uintptr_t)&s_bary[1][0];
  const uint32_t ptf_lds0  = (uint32_t)(uintptr_t)&s_ptf[0][0];
  const uint32_t ptf_lds1  = (uint32_t)(uintptr_t)&s_ptf[1][0];

  const int stride = (int)gridDim.x;
  int tile = (int)blockIdx.x;

  // Prologue: stage first tile into buffer 0 (blockIdx.x < NTILES always).
  stage_tile(t, (uint32_t)tile * TILE, ptf_lds0, bary_lds0, ptf, bary);

  int buf = 0;
  const v4f* __restrict__ attr4 = (const v4f*)attributes;
  v4f* __restrict__ out4 = (v4f*)out;

  for (; tile < NTILES; tile += stride) {
    const int nxt = tile + stride;
    if (nxt < NTILES) {
      // Issue next tile's staging into the other buffer, then wait for the
      // *current* tile's batch (per-wave ASYNCcnt <= 1 leaves only the batch
      // just issued in flight; non-issuing waves pass trivially).
      stage_tile(t, (uint32_t)nxt * TILE,
                 buf ? ptf_lds0 : ptf_lds1,
                 buf ? bary_lds0 : bary_lds1, ptf, bary);
      wait_async_le1();
    } else {
      wait_async_0();
    }
    __syncthreads();   // LDS tile visible to all 8 waves

    // ---- consume tile ----
    const int pix = tile * TILE + p;
    const long long f = s_ptf[buf][p];
    const float w0 = s_bary[buf][p * 3 + 0];
    const float w1 = s_bary[buf][p * 3 + 1];
    const float w2 = s_bary[buf][p * 3 + 2];

    const long long fc = (f < 0) ? 0 : f;          // guard for f == -1
    const long long fb = fc * 3;
    const int i0 = (int)faces[fb + 0];             // L2-resident gathers (RT)
    const int i1 = (int)faces[fb + 1];
    const int i2 = (int)faces[fb + 2];

    // One float4 of the 16-float attribute row per lane -> the 4 lanes of a
    // pixel issue one coalesced 64B row fetch (global_load_b128 each).
    const v4f a0 = attr4[(size_t)i0 * 4 + c];
    const v4f a1 = attr4[(size_t)i1 * 4 + c];
    const v4f a2 = attr4[(size_t)i2 * 4 + c];

    v4f r;
    r.x = fmaf(w2, a2.x, fmaf(w1, a1.x, w0 * a0.x));
    r.y = fmaf(w2, a2.y, fmaf(w1, a1.y, w0 * a0.y));
    r.z = fmaf(w2, a2.z, fmaf(w1, a1.z, w0 * a0.z));
    r.w = fmaf(w2, a2.w, fmaf(w1, a1.w, w0 * a0.w));

    // Write-once streams: non-temporal so the 68MB output stream does not
    // displace the L2-resident faces/attributes tables.
    __builtin_nontemporal_store(r, &out4[(size_t)pix * 4 + c]);
    if (c == 0)
      __builtin_nontemporal_store((f != -1) ? 1.0f : 0.0f, &mask[pix]);

    __syncthreads();   // all reads of buf done before it is restaged
    buf ^= 1;
  }
}

extern "C" void kernel_launch(void* const* d_in, const int* in_sizes, int n_in,
                              void* d_out, int out_size, void* d_ws, size_t ws_size,
                              hipStream_t stream) {
  const float*     attributes = (const float*)d_in[0];      // (V,16) f32
  const float*     bary       = (const float*)d_in[1];      // (H,W,3) f32
  const long long* faces      = (const long long*)d_in[2];  // (F,3) i64
  const long long* ptf        = (const long long*)d_in[3];  // (H,W) i64

  float* out  = (float*)d_out;                              // (H,W,16)
  float* mask = out + (size_t)HW_PIX * D_ATTR;              // (H,W)

  hipLaunchKernelGGL(renderer_interp_kernel, dim3(NBLOCKS), dim3(NTHREADS), 0,
                     stream, attributes, bary, faces, ptf, out, mask);
}